// Graph_Convolution_59957743452553
// MI455X (gfx1250) — compile-verified
//
#include <hip/hip_runtime.h>
#include <hip/hip_bf16.h>

typedef float v2f __attribute__((ext_vector_type(2)));
typedef float v8f __attribute__((ext_vector_type(8)));

#define N_NODES  20000
#define IN_FEAT  256
#define OUT_FEAT 128
#define N_EDGES  640000
#define LDS_STRIDE 260   // 256 + 4 pad: row stride % 64 banks == 4 -> conflict-free column reads

// ---------------------------------------------------------------------------
// Kernel 1: h = x @ W via V_WMMA_F32_16X16X4_F32 (fp32, full accuracy).
// Block = 256 threads = 8 waves. Block handles 16 rows x 128 cols.
// Wave w owns columns [16w, 16w+16). K looped 256/4 = 64 WMMA ops per wave.
// ---------------------------------------------------------------------------
__global__ __launch_bounds__(256) void gcn_gemm_wmma(const float* __restrict__ x,
                                                     const float* __restrict__ w,
                                                     float* __restrict__ h) {
    __shared__ float xs[16 * LDS_STRIDE];

    const int tid  = threadIdx.x;
    const int row0 = blockIdx.x * 16;        // 1250 blocks * 16 = 20000 exactly

    // Cooperative load of the 16x256 x-tile: 1024 float4, 256 threads -> 4 each.
    for (int i = 0; i < 4; ++i) {
        int idx = tid + i * 256;             // float4 index 0..1023
        int r   = idx >> 6;                  // /64 float4 per row
        int c4  = idx & 63;
        float4 v = ((const float4*)(x + (size_t)(row0 + r) * IN_FEAT))[c4];
        float* dst = &xs[r * LDS_STRIDE + c4 * 4];
        dst[0] = v.x; dst[1] = v.y; dst[2] = v.z; dst[3] = v.w;
    }
    __syncthreads();

    const int wave = tid >> 5;               // 0..7 -> column slab
    const int lane = tid & 31;
    const int n0   = wave * 16;
    const int m    = lane & 15;              // M (for A) / N (for B,C) sub-index
    const int kq   = (lane >> 4) * 2;        // K sub-offset: lanes 0-15 -> 0, 16-31 -> 2

    v8f acc = {};                            // 16x16 f32 accumulator (8 VGPRs)

    #pragma unroll 4
    for (int kk = 0; kk < IN_FEAT; kk += 4) {
        // A 16x4 (ISA layout): lane m holds x[row0+m][kk+kq+0 .. +1]
        v2f a, b;
        a.x = xs[m * LDS_STRIDE + kk + kq + 0];
        a.y = xs[m * LDS_STRIDE + kk + kq + 1];
        // B 4x16 (row striped across lanes): VGPR j, this lane-half -> K = kk+kq+j, N = n0+m
        b.x = w[(size_t)(kk + kq + 0) * OUT_FEAT + n0 + m];
        b.y = w[(size_t)(kk + kq + 1) * OUT_FEAT + n0 + m];
        acc = __builtin_amdgcn_wmma_f32_16x16x4_f32(
            /*neg_a=*/false, a, /*neg_b=*/false, b,
            /*c_mod=*/(short)0, acc, /*reuse_a=*/false, /*reuse_b=*/false);
    }

    // C/D 16x16 layout: VGPR v, lanes 0-15 -> M=v, lanes 16-31 -> M=v+8; N = lane&15.
    const int mbase = (lane >> 4) * 8;
    const int cn    = n0 + m;
    const float* accp = (const float*)&acc;
    #pragma unroll
    for (int v = 0; v < 8; ++v) {
        h[(size_t)(row0 + mbase + v) * OUT_FEAT + cn] = accp[v];
    }
}

// ---------------------------------------------------------------------------
// Kernel 2: zero the output accumulator (d_out is poisoned by the harness).
// ---------------------------------------------------------------------------
__global__ __launch_bounds__(256) void gcn_zero(float4* __restrict__ out, int n4) {
    int i = blockIdx.x * blockDim.x + threadIdx.x;
    if (i < n4) out[i] = make_float4(0.f, 0.f, 0.f, 0.f);
}

// ---------------------------------------------------------------------------
// Kernel 3: edge scatter-add. One wave32 per edge: lane l gathers
// h[col][4l..4l+3] as a float4 and atomically adds into out[row].
// h and out are L2-resident (10 MB each in 192 MB L2) -> L2 atomic bound.
// ---------------------------------------------------------------------------
__global__ __launch_bounds__(256) void gcn_scatter(const long long* __restrict__ ei,
                                                   const float* __restrict__ h,
                                                   float* __restrict__ out) {
    const int e = blockIdx.x * 8 + (threadIdx.x >> 5);   // 8 waves per block
    if (e >= N_EDGES) return;
    const int lane = threadIdx.x & 31;

    const long long row = ei[e];             // destination node
    const long long col = ei[N_EDGES + e];   // source node

    const float4 v = ((const float4*)(h + (size_t)col * OUT_FEAT))[lane];
    float* dst = out + (size_t)row * OUT_FEAT + lane * 4;
    atomicAdd(dst + 0, v.x);
    atomicAdd(dst + 1, v.y);
    atomicAdd(dst + 2, v.z);
    atomicAdd(dst + 3, v.w);
}

// ---------------------------------------------------------------------------
// Kernel 4: out = relu(out + bias), in place, float4 per thread.
// ---------------------------------------------------------------------------
__global__ __launch_bounds__(256) void gcn_bias_relu(float* __restrict__ out,
                                                     const float* __restrict__ bias) {
    const int i    = blockIdx.x * blockDim.x + threadIdx.x;   // float4 index
    const int base = i * 4;
    if (base < N_NODES * OUT_FEAT) {
        const int f = base & (OUT_FEAT - 1);
        float4 v = *(const float4*)(out + base);
        v.x = fmaxf(v.x + bias[f + 0], 0.f);
        v.y = fmaxf(v.y + bias[f + 1], 0.f);
        v.z = fmaxf(v.z + bias[f + 2], 0.f);
        v.w = fmaxf(v.w + bias[f + 3], 0.f);
        *(float4*)(out + base) = v;
    }
}

// ---------------------------------------------------------------------------
extern "C" void kernel_launch(void* const* d_in, const int* in_sizes, int n_in,
                              void* d_out, int out_size, void* d_ws, size_t ws_size,
                              hipStream_t stream) {
    const float*     x    = (const float*)d_in[0];
    const long long* ei   = (const long long*)d_in[1];   // int64 edge_index [2, N_EDGES]
    const float*     w    = (const float*)d_in[2];
    const float*     bias = (const float*)d_in[3];
    float*           out  = (float*)d_out;
    float*           h    = (float*)d_ws;                // N_NODES*OUT_FEAT*4 = 10.24 MB

    (void)in_sizes; (void)n_in; (void)out_size; (void)ws_size;

    // 1) h = x @ W  (fp32 WMMA)
    gcn_gemm_wmma<<<N_NODES / 16, 256, 0, stream>>>(x, w, h);

    // 2) out = 0
    const int n4 = (N_NODES * OUT_FEAT) / 4;             // 640000 float4
    gcn_zero<<<(n4 + 255) / 256, 256, 0, stream>>>((float4*)out, n4);

    // 3) scatter-add along edges (wave per edge)
    gcn_scatter<<<N_EDGES / 8, 256, 0, stream>>>(ei, h, out);

    // 4) out = relu(out + bias)
    gcn_bias_relu<<<(n4 + 255) / 256, 256, 0, stream>>>(out, bias);
}